// OrthogonalLoss_41223096107180
// MI455X (gfx1250) — compile-verified
//
#include <hip/hip_runtime.h>

// Problem constants (fixed by the reference setup_inputs()).
#define BATCH        16
#define NROWS        16384
#define DIM          128
#define NUM_CLUSTERS 32

#define BLOCKS_PER_BATCH 64
#define ROWS_PER_BLOCK   (NROWS / BLOCKS_PER_BATCH)   // 256
#define MEANS_STRIDE     (DIM + 4)                    // 132: LDS bank-conflict padding

typedef __attribute__((ext_vector_type(2))) float v2f;
typedef __attribute__((ext_vector_type(8))) float v8f;

// ---------------------------------------------------------------------------
// Kernel 0: zero the workspace accumulators (ws is poisoned, not zeroed).
// ---------------------------------------------------------------------------
__global__ void ol_zero_ws(float* __restrict__ ws, int n) {
    int i = blockIdx.x * blockDim.x + threadIdx.x;
    if (i < n) ws[i] = 0.0f;
}

// ---------------------------------------------------------------------------
// Kernel 1: segment sum + counts. Grid = (BLOCKS_PER_BATCH, BATCH), 256 thr.
// One wave handles one row at a time: lane l loads x[row][4l..4l+3] as a
// single b128 load (512B contiguous per wave) and ds_add_f32's it into the
// LDS accumulator row selected by the cluster id. Block flushes once with
// global atomics.
// ---------------------------------------------------------------------------
__global__ __launch_bounds__(256) void ol_segsum(const float* __restrict__ x,
                                                 const int*   __restrict__ cids,
                                                 float* __restrict__ wsSums,
                                                 float* __restrict__ wsCounts) {
    __shared__ float sAcc[NUM_CLUSTERS * DIM];   // 16 KB
    __shared__ float sCnt[NUM_CLUSTERS];

    const int b   = blockIdx.y;
    const int tid = threadIdx.x;

    for (int i = tid; i < NUM_CLUSTERS * DIM; i += 256) sAcc[i] = 0.0f;
    if (tid < NUM_CLUSTERS) sCnt[tid] = 0.0f;
    __syncthreads();

    const int lane = tid & 31;
    const int wave = tid >> 5;             // 8 waves
    const int rowBase = blockIdx.x * ROWS_PER_BLOCK;

    const float* xb = x    + (size_t)b * NROWS * DIM;
    const int*   cb = cids + (size_t)b * NROWS;

    for (int r = rowBase + wave; r < rowBase + ROWS_PER_BLOCK; r += 8) {
        const int c = cb[r];  // wave-uniform load
        const float4 v = *(const float4*)(xb + (size_t)r * DIM + lane * 4);
        float* dst = &sAcc[c * DIM + lane * 4];
        atomicAdd(dst + 0, v.x);
        atomicAdd(dst + 1, v.y);
        atomicAdd(dst + 2, v.z);
        atomicAdd(dst + 3, v.w);
        if (lane == 0) atomicAdd(&sCnt[c], 1.0f);
    }
    __syncthreads();

    float* gs = wsSums + (size_t)b * NUM_CLUSTERS * DIM;
    for (int i = tid; i < NUM_CLUSTERS * DIM; i += 256) atomicAdd(&gs[i], sAcc[i]);
    if (tid < NUM_CLUSTERS) atomicAdd(&wsCounts[b * NUM_CLUSTERS + tid], sCnt[tid]);
}

// ---------------------------------------------------------------------------
// Kernel 2: per-batch Gram matrix of cluster means via V_WMMA_F32_16X16X4_F32.
// One block per batch, 128 threads = 4 waves; wave w owns the 16x16 output
// tile (tr, tc) = (w>>1, w&1) of the 32x32 Gram. K = 128 in 32 WMMA steps.
//
// f32 A 16x4 layout: lane L holds row M = L%16; VGPR0/1 = K pair selected by
// lane half (L<16 -> K {k0,k0+1}, L>=16 -> K {k0+2,k0+3}). B = A-layout of
// means rows for the column tile (Gram: B-tile = means[colTile]^T).
// C/D 16x16 f32: lane L, VGPR v -> row = v + 8*(L/16), col = L%16.
// ---------------------------------------------------------------------------
__global__ __launch_bounds__(128) void ol_gram_wmma(const float* __restrict__ wsSums,
                                                    const float* __restrict__ wsCounts,
                                                    float* __restrict__ wsLoss) {
    __shared__ float sMeans[NUM_CLUSTERS * MEANS_STRIDE];  // padded stride 132
    __shared__ float sCnt[NUM_CLUSTERS];
    __shared__ float sRed[128];

    const int b   = blockIdx.x;
    const int tid = threadIdx.x;

    if (tid < NUM_CLUSTERS) sCnt[tid] = wsCounts[b * NUM_CLUSTERS + tid];
    __syncthreads();

    const float* gs = wsSums + (size_t)b * NUM_CLUSTERS * DIM;
    for (int i = tid; i < NUM_CLUSTERS * DIM; i += 128) {
        const int c = i >> 7;          // / DIM
        const int d = i & (DIM - 1);   // % DIM
        sMeans[c * MEANS_STRIDE + d] = gs[i] / fmaxf(sCnt[c], 1.0f);
    }
    __syncthreads();

    const int lane = tid & 31;
    const int wave = tid >> 5;
    const int tr = wave >> 1, tc = wave & 1;
    const int m  = lane & 15;          // M (for A) / N (for B) within tile
    const int kh = lane >> 4;          // lane-half selects K pair

    const float* aRow = &sMeans[(tr * 16 + m) * MEANS_STRIDE];
    const float* bRow = &sMeans[(tc * 16 + m) * MEANS_STRIDE];

    v8f acc = {};
    for (int k0 = 0; k0 < DIM; k0 += 4) {
        v2f a, bb;
        a.x  = aRow[k0 + 2 * kh];
        a.y  = aRow[k0 + 2 * kh + 1];
        bb.x = bRow[k0 + 2 * kh];
        bb.y = bRow[k0 + 2 * kh + 1];
        // 8 args: (neg_a, A, neg_b, B, c_mod, C, reuse_a, reuse_b)
        acc = __builtin_amdgcn_wmma_f32_16x16x4_f32(false, a, false, bb,
                                                    (short)0, acc, false, false);
    }

    // Strict upper-triangular |.| partial sum from the C/D register layout.
    float part = 0.0f;
    const int col     = tc * 16 + m;
    const int rowBase = tr * 16 + 8 * kh;
#pragma unroll
    for (int v = 0; v < 8; ++v) {
        const int row = rowBase + v;
        if (col > row) part += fabsf(acc[v]);
    }
    sRed[tid] = part;
    __syncthreads();

    if (tid == 0) {
        float s = 0.0f;
        for (int i = 0; i < 128; ++i) s += sRed[i];
        float valid = 0.0f;
        for (int c = 0; c < NUM_CLUSTERS; ++c) valid += (sCnt[c] > 0.0f) ? 1.0f : 0.0f;
        const float denom = (valid > 1.0f) ? valid * (valid - 1.0f) * 0.5f : 1.0f;
        wsLoss[b] = s / denom;
    }
}

// ---------------------------------------------------------------------------
// Kernel 3: mean over batches -> scalar output.
// ---------------------------------------------------------------------------
__global__ void ol_finalize(const float* __restrict__ wsLoss, float* __restrict__ out) {
    if (threadIdx.x == 0) {
        float s = 0.0f;
        for (int b = 0; b < BATCH; ++b) s += wsLoss[b];
        out[0] = s / (float)BATCH;
    }
}

// ---------------------------------------------------------------------------
// Workspace layout (floats):
//   [0,                B*C*D)        segment sums
//   [B*C*D,            B*C*D + B*C)  counts
//   [B*C*D + B*C,      ... + B)      per-batch loss
// Total: 66,064 floats (~264 KB).
// ---------------------------------------------------------------------------
extern "C" void kernel_launch(void* const* d_in, const int* in_sizes, int n_in,
                              void* d_out, int out_size, void* d_ws, size_t ws_size,
                              hipStream_t stream) {
    (void)in_sizes; (void)n_in; (void)out_size; (void)ws_size;

    const float* x    = (const float*)d_in[0];
    const int*   cids = (const int*)d_in[1];
    float*       out  = (float*)d_out;

    float* ws       = (float*)d_ws;
    float* wsSums   = ws;
    float* wsCounts = ws + (size_t)BATCH * NUM_CLUSTERS * DIM;
    float* wsLoss   = wsCounts + BATCH * NUM_CLUSTERS;

    const int nZero = BATCH * NUM_CLUSTERS * DIM + BATCH * NUM_CLUSTERS + BATCH;
    ol_zero_ws<<<(nZero + 255) / 256, 256, 0, stream>>>(ws, nZero);

    ol_segsum<<<dim3(BLOCKS_PER_BATCH, BATCH), 256, 0, stream>>>(x, cids, wsSums, wsCounts);

    ol_gram_wmma<<<BATCH, 128, 0, stream>>>(wsSums, wsCounts, wsLoss);

    ol_finalize<<<1, 32, 0, stream>>>(wsLoss, out);
}